// GCN_34411277976330
// MI455X (gfx1250) — compile-verified
//
#include <hip/hip_runtime.h>
#include <hip/hip_bf16.h>

typedef __attribute__((ext_vector_type(2))) float v2f;
typedef __attribute__((ext_vector_type(8))) float v8f;

// ---------------------------------------------------------------------------
// Small init kernels
// ---------------------------------------------------------------------------
__global__ void init_deg_kernel(float* __restrict__ deg, int N) {
    int i = blockIdx.x * blockDim.x + threadIdx.x;
    if (i < N) deg[i] = 1.0f;   // self-loop contributes 1 to degree
}

__global__ void zero_kernel(float* __restrict__ p, int n) {
    int i = blockIdx.x * blockDim.x + threadIdx.x;
    if (i < n) p[i] = 0.0f;
}

// deg -> dinv (in place). deg >= 1 always (self-loops), guard kept to match ref.
__global__ void rsqrt_kernel(float* __restrict__ deg, int N) {
    int i = blockIdx.x * blockDim.x + threadIdx.x;
    if (i < N) {
        float d = deg[i];
        deg[i] = (d > 0.0f) ? rsqrtf(d) : 0.0f;
    }
}

// ---------------------------------------------------------------------------
// Degree accumulation over edge targets
// ---------------------------------------------------------------------------
__global__ void degree_kernel(const int* __restrict__ col, float* __restrict__ deg,
                              long long E) {
    long long e = (long long)blockIdx.x * blockDim.x + threadIdx.x;
    long long stride = (long long)gridDim.x * blockDim.x;
    for (; e < E; e += stride) {
        unsafeAtomicAdd(&deg[col[e]], 1.0f);
    }
}

// ---------------------------------------------------------------------------
// conv1 dense transform: xw1 = x @ W1   ([N,3] @ [3,8], K padded to 4, N to 16)
// One wave per 16-node tile, single V_WMMA_F32_16X16X4_F32.
// ---------------------------------------------------------------------------
__global__ void xw1_wmma_kernel(const float* __restrict__ x,
                                const float* __restrict__ W1,
                                float* __restrict__ xw1, int N) {
    int wave = (blockIdx.x * blockDim.x + threadIdx.x) >> 5;
    int lane = threadIdx.x & 31;
    int half = lane >> 4;     // 0: K={0,1}, D rows 0..7 ; 1: K={2,3}, D rows 8..15
    int l    = lane & 15;
    int base = wave * 16;
    if (base >= N) return;    // wave-uniform exit: EXEC all-ones at the WMMA

    bool full = (base + 16 <= N);                 // wave-uniform fast path
    int node  = base + l;
    if (!full && node >= N) node = N - 1;         // clamp loads on the last tile

    // B: unconditional load from clamped column, masked to zero for n>=8 / K=3 pad
    int   ln   = l & 7;
    float mask = (l < 8) ? 1.0f : 0.0f;
    v2f a, b;
    if (half == 0) {
        a.x = x[node * 3 + 0];
        a.y = x[node * 3 + 1];
        b.x = W1[0 * 8 + ln] * mask;
        b.y = W1[1 * 8 + ln] * mask;
    } else {
        a.x = x[node * 3 + 2];
        a.y = 0.0f;                                // K=3 padding
        b.x = W1[2 * 8 + ln] * mask;
        b.y = 0.0f;
    }

    v8f c = {};
    c = __builtin_amdgcn_wmma_f32_16x16x4_f32(false, a, false, b,
                                              (short)0, c, false, false);

    // D VGPR r: lanes0-15 -> M=r, lanes16-31 -> M=r+8; lane l holds column l.
    float* dst = xw1 + ((long long)(base + half * 8)) * 8 + l;
    if (full) {
        if (l < 8) {
            #pragma unroll
            for (int r = 0; r < 8; ++r) dst[r * 8] = c[r];
        }
    } else {
        if (l < 8) {
            #pragma unroll
            for (int r = 0; r < 8; ++r) {
                int row = base + half * 8 + r;
                if (row < N) xw1[(long long)row * 8 + l] = c[r];
            }
        }
    }
}

// ---------------------------------------------------------------------------
// conv2 dense transform, fused: xw2 = relu(h1 + b1) @ W2   ([N,8] @ [8,8])
// Two accumulated 16x16x4 WMMAs (K blocks 0..3 and 4..7) per 16-node tile.
// ---------------------------------------------------------------------------
__global__ void xw2_wmma_kernel(const float* __restrict__ h1,
                                const float* __restrict__ b1,
                                const float* __restrict__ W2,
                                float* __restrict__ xw2, int N) {
    int wave = (blockIdx.x * blockDim.x + threadIdx.x) >> 5;
    int lane = threadIdx.x & 31;
    int half = lane >> 4;
    int l    = lane & 15;
    int base = wave * 16;
    if (base >= N) return;

    bool full = (base + 16 <= N);
    int node  = base + l;
    if (!full && node >= N) node = N - 1;

    int   ln   = l & 7;
    float mask = (l < 8) ? 1.0f : 0.0f;
    const v2f* hrow = (const v2f*)(h1 + (long long)node * 8);   // 4 x float2 per node

    v8f c = {};
    #pragma unroll
    for (int kb = 0; kb < 2; ++kb) {
        int k0 = kb * 4 + half * 2;          // this lane supplies K = k0, k0+1 (k0 even)
        v2f hv = hrow[k0 >> 1];              // single b64 load
        v2f a, b;
        a.x = fmaxf(hv.x + b1[k0],     0.0f);
        a.y = fmaxf(hv.y + b1[k0 + 1], 0.0f);
        b.x = W2[(k0)     * 8 + ln] * mask;
        b.y = W2[(k0 + 1) * 8 + ln] * mask;
        c = __builtin_amdgcn_wmma_f32_16x16x4_f32(false, a, false, b,
                                                  (short)0, c, false, false);
    }

    float* dst = xw2 + ((long long)(base + half * 8)) * 8 + l;
    if (full) {
        if (l < 8) {
            #pragma unroll
            for (int r = 0; r < 8; ++r) dst[r * 8] = c[r];
        }
    } else {
        if (l < 8) {
            #pragma unroll
            for (int r = 0; r < 8; ++r) {
                int row = base + half * 8 + r;
                if (row < N) xw2[(long long)row * 8 + l] = c[r];
            }
        }
    }
}

// ---------------------------------------------------------------------------
// Self-loop contribution (overwrite-init of the accumulator):
//   acc[i,:] = xw[i,:] * dinv[i]^2
// ---------------------------------------------------------------------------
__global__ void selfloop_init_kernel(const float* __restrict__ xw,
                                     const float* __restrict__ dinv,
                                     float* __restrict__ acc, int N) {
    int i = blockIdx.x * blockDim.x + threadIdx.x;
    if (i >= N) return;
    float w = dinv[i] * dinv[i];
    const float4* s = (const float4*)(xw + (long long)i * 8);
    float4* d = (float4*)(acc + (long long)i * 8);
    float4 s0 = s[0], s1 = s[1];
    d[0] = make_float4(s0.x * w, s0.y * w, s0.z * w, s0.w * w);
    d[1] = make_float4(s1.x * w, s1.y * w, s1.z * w, s1.w * w);
}

// ---------------------------------------------------------------------------
// Edge scatter: acc[col[e],:] += xw[row[e],:] * dinv[row]*dinv[col]
// 8 f32 hardware atomics per edge; accumulators are L2-resident (32MB << 192MB).
// ---------------------------------------------------------------------------
__global__ void edge_scatter_kernel(const int* __restrict__ row,
                                    const int* __restrict__ col,
                                    const float* __restrict__ dinv,
                                    const float* __restrict__ xw,
                                    float* __restrict__ acc, long long E) {
    long long e = (long long)blockIdx.x * blockDim.x + threadIdx.x;
    long long stride = (long long)gridDim.x * blockDim.x;
    for (; e < E; e += stride) {
        int r = row[e];
        int c = col[e];
        float w = dinv[r] * dinv[c];
        const float4* s = (const float4*)(xw + (long long)r * 8);
        float4 m0 = s[0], m1 = s[1];
        float* d = acc + (long long)c * 8;
        unsafeAtomicAdd(d + 0, m0.x * w);
        unsafeAtomicAdd(d + 1, m0.y * w);
        unsafeAtomicAdd(d + 2, m0.z * w);
        unsafeAtomicAdd(d + 3, m0.w * w);
        unsafeAtomicAdd(d + 4, m1.x * w);
        unsafeAtomicAdd(d + 5, m1.y * w);
        unsafeAtomicAdd(d + 6, m1.z * w);
        unsafeAtomicAdd(d + 7, m1.w * w);
    }
}

// ---------------------------------------------------------------------------
// Global mean pool (accumulate phase): relu(h2 + b2) summed per graph + counts
// ---------------------------------------------------------------------------
__global__ void pool_kernel(const float* __restrict__ h2,
                            const float* __restrict__ b2,
                            const int* __restrict__ batch,
                            float* __restrict__ psum,
                            float* __restrict__ pcnt, int N) {
    int i = blockIdx.x * blockDim.x + threadIdx.x;
    if (i >= N) return;
    int g = batch[i];
    unsafeAtomicAdd(&pcnt[g], 1.0f);
    float* d = psum + (long long)g * 8;
    #pragma unroll
    for (int f = 0; f < 8; ++f) {
        float v = fmaxf(h2[(long long)i * 8 + f] + b2[f], 0.0f);
        unsafeAtomicAdd(d + f, v);
    }
}

// ---------------------------------------------------------------------------
// Final: out[g,:] = (psum[g,:]/max(cnt,1)) @ Wl + bl      ([1024,8]@[8,2])
// ---------------------------------------------------------------------------
__global__ void out_kernel(const float* __restrict__ psum,
                           const float* __restrict__ pcnt,
                           const float* __restrict__ Wl,
                           const float* __restrict__ bl,
                           float* __restrict__ out, int G) {
    int g = blockIdx.x * blockDim.x + threadIdx.x;
    if (g >= G) return;
    float inv = 1.0f / fmaxf(pcnt[g], 1.0f);
    float o0 = bl[0], o1 = bl[1];
    #pragma unroll
    for (int f = 0; f < 8; ++f) {
        float v = psum[(long long)g * 8 + f] * inv;
        o0 += v * Wl[f * 2 + 0];
        o1 += v * Wl[f * 2 + 1];
    }
    out[g * 2 + 0] = o0;
    out[g * 2 + 1] = o1;
}

// ---------------------------------------------------------------------------
// Host orchestration
// ---------------------------------------------------------------------------
extern "C" void kernel_launch(void* const* d_in, const int* in_sizes, int n_in,
                              void* d_out, int out_size, void* d_ws, size_t ws_size,
                              hipStream_t stream) {
    const float* x     = (const float*)d_in[0];
    const int*   eidx  = (const int*)d_in[1];
    const int*   batch = (const int*)d_in[2];
    const float* W1    = (const float*)d_in[3];
    const float* b1    = (const float*)d_in[4];
    const float* W2    = (const float*)d_in[5];
    const float* b2    = (const float*)d_in[6];
    const float* Wl    = (const float*)d_in[7];
    const float* bl    = (const float*)d_in[8];
    float* out = (float*)d_out;

    const int       N = in_sizes[0] / 3;
    const long long E = (long long)in_sizes[1] / 2;
    const int       G = out_size / 2;

    const int* erow = eidx;          // edge_index[0] : sources
    const int* ecol = eidx + E;      // edge_index[1] : targets

    // Workspace layout (256B-aligned slabs). bufA holds xw1 then xw2;
    // bufB holds the conv1 accumulator (h1) then the conv2 accumulator (h2).
    float* ws = (float*)d_ws;
    size_t off = 0;
    float* dinv = ws + off;  off += (size_t)N;            off = (off + 63) & ~(size_t)63;
    float* bufA = ws + off;  off += (size_t)N * 8;        off = (off + 63) & ~(size_t)63;
    float* bufB = ws + off;  off += (size_t)N * 8;        off = (off + 63) & ~(size_t)63;
    float* psum = ws + off;  off += (size_t)G * 8;
    float* pcnt = ws + off;  off += (size_t)G;

    const int T = 256;
    const int nodeBlocks = (N + T - 1) / T;
    const int waveBlocks = (N + 16 * 8 - 1) / (16 * 8);   // 8 waves/block, 16 nodes/wave
    const int edgeBlocks = 16384;                          // grid-stride over 10M edges
    const int poolInit   = (G * 9 + T - 1) / T;

    // 1) init: deg = 1 (self-loop), pooled sums/counts = 0
    init_deg_kernel<<<nodeBlocks, T, 0, stream>>>(dinv, N);
    zero_kernel<<<poolInit, T, 0, stream>>>(psum, G * 9);  // psum (8G) + pcnt (G) contiguous

    // 2) degree over edge targets, then dinv = rsqrt(deg) in place
    degree_kernel<<<edgeBlocks, T, 0, stream>>>(ecol, dinv, E);
    rsqrt_kernel<<<nodeBlocks, T, 0, stream>>>(dinv, N);

    // 3) conv1 dense transform (WMMA f32 16x16x4): bufA = x @ W1
    xw1_wmma_kernel<<<waveBlocks, T, 0, stream>>>(x, W1, bufA, N);

    // 4) conv1 aggregation: bufB = self-loop term, then edge scatter-add
    selfloop_init_kernel<<<nodeBlocks, T, 0, stream>>>(bufA, dinv, bufB, N);
    edge_scatter_kernel<<<edgeBlocks, T, 0, stream>>>(erow, ecol, dinv, bufA, bufB, E);

    // 5) conv2 dense transform, fused bias+relu (WMMA x2): bufA = relu(bufB+b1) @ W2
    xw2_wmma_kernel<<<waveBlocks, T, 0, stream>>>(bufB, b1, W2, bufA, N);

    // 6) conv2 aggregation into bufB (h1 is dead)
    selfloop_init_kernel<<<nodeBlocks, T, 0, stream>>>(bufA, dinv, bufB, N);
    edge_scatter_kernel<<<edgeBlocks, T, 0, stream>>>(erow, ecol, dinv, bufA, bufB, E);

    // 7) global mean pool accumulate: relu(bufB + b2) -> psum/pcnt
    pool_kernel<<<nodeBlocks, T, 0, stream>>>(bufB, b2, batch, psum, pcnt, N);

    // 8) final linear
    out_kernel<<<(G + T - 1) / T, T, 0, stream>>>(psum, pcnt, Wl, bl, out, G);
}